// ADQC_basic_41532333752433
// MI455X (gfx1250) — compile-verified
//
#include <hip/hip_runtime.h>
#include <hip/hip_bf16.h>

// ---------------------------------------------------------------------------
// 23-qubit brickwall circuit, 33 two-qubit gates.
// Strategy:
//   1) copy state (2^23 complex64 = 64MB, fits in 192MB L2) into d_out
//   2) tiny kernel: Newton-Schulz polar decomposition of each 4x4 latent
//      -> unitary U = u @ vh; store as real 8x8 matrix in d_ws
//   3) 33 sequential in-place gate kernels, each using chained
//      V_WMMA_F32_16X16X4_F32 ops: D(16x16) = A(16x4)xB(4x16)+C with
//      A = real 8x8 gate matrix (K split over 2 accumulating WMMAs),
//      B columns = 16 state groups, D rows = output components.
// ---------------------------------------------------------------------------

typedef __attribute__((ext_vector_type(2))) float v2f;
typedef __attribute__((ext_vector_type(8))) float v8f;

#define N_QUBIT 23
#define N_GATES 33
#define N_GROUPS (1u << (N_QUBIT - 2))   // 2^21 groups of 4 amplitudes
#define N_TILES  (N_GROUPS / 16u)        // 131072 tiles of 16 groups

// ---------------------------------------------------------------------------
// Phase 1: polar factor of each 4x4 complex latent via Newton-Schulz:
//   X0 = g / ||g||_F ;  X <- 1.5*X - 0.5*X*(X^H X)   (converges to u @ vh)
// Output: real 8x8 matrix Mreal row-major, Mreal acts on (re0,im0,...,re3,im3).
// ---------------------------------------------------------------------------
__global__ void polar_gates_kernel(const float* __restrict__ gp,
                                   float* __restrict__ Mout) {
  int i = blockIdx.x * blockDim.x + threadIdx.x;
  if (i >= N_GATES) return;

  float xr[4][4], xi[4][4];
  const float* p = gp + i * 32;          // 16 complex = 32 floats
  float nrm = 1e-30f;
  for (int r = 0; r < 4; ++r)
    for (int c = 0; c < 4; ++c) {
      float re = p[(r * 4 + c) * 2 + 0];
      float im = p[(r * 4 + c) * 2 + 1];
      xr[r][c] = re; xi[r][c] = im;
      nrm += re * re + im * im;
    }
  float s = rsqrtf(nrm);
  for (int r = 0; r < 4; ++r)
    for (int c = 0; c < 4; ++c) { xr[r][c] *= s; xi[r][c] *= s; }

  for (int it = 0; it < 48; ++it) {
    // Y = X^H X  (Y[a][b] = sum_k conj(X[k][a]) * X[k][b])
    float yr[4][4], yi[4][4];
    for (int a = 0; a < 4; ++a)
      for (int b = 0; b < 4; ++b) {
        float sr = 0.f, si = 0.f;
        for (int k = 0; k < 4; ++k) {
          sr += xr[k][a] * xr[k][b] + xi[k][a] * xi[k][b];
          si += xr[k][a] * xi[k][b] - xi[k][a] * xr[k][b];
        }
        yr[a][b] = sr; yi[a][b] = si;
      }
    // Z = 3I - Y ;  X' = 0.5 * X * Z
    float nr[4][4], ni[4][4];
    for (int a = 0; a < 4; ++a)
      for (int b = 0; b < 4; ++b) {
        float sr = 0.f, si = 0.f;
        for (int k = 0; k < 4; ++k) {
          float zr = (k == b ? 3.f : 0.f) - yr[k][b];
          float zi = -yi[k][b];
          sr += xr[a][k] * zr - xi[a][k] * zi;
          si += xr[a][k] * zi + xi[a][k] * zr;
        }
        nr[a][b] = 0.5f * sr; ni[a][b] = 0.5f * si;
      }
    for (int a = 0; a < 4; ++a)
      for (int b = 0; b < 4; ++b) { xr[a][b] = nr[a][b]; xi[a][b] = ni[a][b]; }
  }

  // complex 4x4 -> real 8x8:  [Re -Im; Im Re] interleaved per component
  float* o = Mout + i * 64;
  for (int r = 0; r < 4; ++r)
    for (int c = 0; c < 4; ++c) {
      float re = xr[r][c], im = xi[r][c];
      o[(2 * r + 0) * 8 + 2 * c + 0] = re;
      o[(2 * r + 0) * 8 + 2 * c + 1] = -im;
      o[(2 * r + 1) * 8 + 2 * c + 0] = im;
      o[(2 * r + 1) * 8 + 2 * c + 1] = re;
    }
}

// ---------------------------------------------------------------------------
// Phase 2: apply one gate in-place with V_WMMA_F32_16X16X4_F32.
//   m = 2^(21-q) : complex-element stride of qubit q+1; qubit q stride = 2m.
//   group g -> amp base b = ((g & ~(m-1)) << 2) | (g & (m-1)); amps at b+i*m.
//   Per tile of 16 groups: lane n<16 carries group n, lane n+16 the K=2,3 rows.
// ---------------------------------------------------------------------------
__global__ __launch_bounds__(256) void apply_gate_wmma_kernel(
    float* __restrict__ state, const float* __restrict__ M8, int m) {
  const int lane = threadIdx.x & 31;
  const int half = lane >> 4;            // 0: K=0,1   1: K=2,3
  const int n    = lane & 15;            // group-in-tile (B,N) / row (A,M)
  const int wave   = (blockIdx.x * blockDim.x + threadIdx.x) >> 5;
  const int nwaves = (gridDim.x * blockDim.x) >> 5;

  // A operands: rows 0..7 = real 8x8 gate matrix, rows 8..15 = 0.
  v2f a1 = {0.f, 0.f}, a2 = {0.f, 0.f};
  if (n < 8) {
    const v2f* Mv = (const v2f*)M8;      // row-major 8x8 as 4 v2f per row
    a1 = Mv[n * 4 + half];               // cols {0,1} / {2,3}
    a2 = Mv[n * 4 + 2 + half];           // cols {4,5} / {6,7}
  }

  v2f* S = (v2f*)state;                  // complex64 viewed as float2
  const unsigned mm1 = (unsigned)m - 1u;
  const unsigned um  = (unsigned)m;

  for (unsigned t = (unsigned)wave; t < N_TILES; t += (unsigned)nwaves) {
    const unsigned g = t * 16u + (unsigned)n;
    const unsigned b = ((g & ~mm1) << 2) | (g & mm1);

    // B: lane n -> amp0 of group n, lane n+16 -> amp1 (then amp2/amp3)
    v2f b1 = S[b + (unsigned)half * um];
    v2f b2 = S[b + (unsigned)(2 + half) * um];

    v8f c = {};
    c = __builtin_amdgcn_wmma_f32_16x16x4_f32(false, a1, false, b1,
                                              (short)0, c, false, false);
    c = __builtin_amdgcn_wmma_f32_16x16x4_f32(false, a2, false, b2,
                                              (short)0, c, false, false);

    // D: vgpr r, lanes 0-15 = component r of group=lane. Store per-group.
    if (half == 0) {
      v2f p;
      p[0] = c[0]; p[1] = c[1]; S[b]            = p;
      p[0] = c[2]; p[1] = c[3]; S[b + um]       = p;
      p[0] = c[4]; p[1] = c[5]; S[b + 2u * um]  = p;
      p[0] = c[6]; p[1] = c[7]; S[b + 3u * um]  = p;
    }
  }
}

// ---------------------------------------------------------------------------
extern "C" void kernel_launch(void* const* d_in, const int* in_sizes, int n_in,
                              void* d_out, int out_size, void* d_ws, size_t ws_size,
                              hipStream_t stream) {
  const float* state_in = (const float*)d_in[0];   // 2^23 complex64
  const float* gp       = (const float*)d_in[1];   // 33 x 4 x 4 complex64
  float* state = (float*)d_out;
  float* Mws   = (float*)d_ws;                     // 33 * 64 floats = 8448 B

  (void)in_sizes; (void)n_in; (void)out_size; (void)ws_size;

  // Copy input state; all gates then applied in place on d_out (fits in L2).
  hipMemcpyAsync(state, state_in, (size_t)8u << 23, hipMemcpyDeviceToDevice,
                 stream);

  polar_gates_kernel<<<1, 64, 0, stream>>>(gp, Mws);

  // Brickwall positions: layer 0/2: q = 0,2,...,20 ; layer 1: q = 1,3,...,21
  int qs[N_GATES]; int k = 0;
  for (int l = 0; l < 3; ++l) {
    for (int q = (l & 1); q < N_QUBIT - 1; q += 2) qs[k++] = q;
  }

  for (int i = 0; i < N_GATES; ++i) {
    const int m = 1 << (21 - qs[i]);     // stride of qubit q+1 in elements
    apply_gate_wmma_kernel<<<512, 256, 0, stream>>>(state, Mws + i * 64, m);
  }
}